// GIND_2989297238465
// MI455X (gfx1250) — compile-verified
//
#include <hip/hip_runtime.h>
#include <hip/hip_bf16.h>
#include <math.h>

#define HCH   64
#define INCH  128
#define OUTCH 40
#define ALPHA 0.8f
#define LN_EPS 1e-5f

typedef float v2f __attribute__((ext_vector_type(2)));
typedef float v8f __attribute__((ext_vector_type(8)));

// ---------------------------------------------------------------------------
// WMMA helpers: 16x16 f32 tile accumulate with V_WMMA_F32_16X16X4_F32.
// A is row-major [M, lda]; "bt" variant takes B stored as [Ncols, ldb] and
// computes A @ B^T (PyTorch nn.Linear weight layout); "bn" takes row-major
// B [K, ldb] and computes A @ B.
// ---------------------------------------------------------------------------
__device__ __forceinline__ v8f wmma_acc_bt(const float* __restrict__ A, int lda,
                                           const float* __restrict__ Bt, int ldb,
                                           int K, int m, int n, int half, v8f acc) {
  for (int k0 = 0; k0 < K; k0 += 4) {
    const int ka = k0 + half * 2;
    v2f av, bv;
    av.x = A[m * lda + ka];   av.y = A[m * lda + ka + 1];
    bv.x = Bt[n * ldb + ka];  bv.y = Bt[n * ldb + ka + 1];
    acc = __builtin_amdgcn_wmma_f32_16x16x4_f32(false, av, false, bv,
                                                (short)0, acc, false, false);
  }
  return acc;
}

__device__ __forceinline__ v8f wmma_acc_bn(const float* __restrict__ A, int lda,
                                           const float* __restrict__ B, int ldb,
                                           int K, int m, int n, int half, v8f acc) {
  for (int k0 = 0; k0 < K; k0 += 4) {
    const int ka = k0 + half * 2;
    v2f av, bv;
    av.x = A[m * lda + ka];       av.y = A[m * lda + ka + 1];
    bv.x = B[ka * ldb + n];       bv.y = B[(ka + 1) * ldb + n];
    acc = __builtin_amdgcn_wmma_f32_16x16x4_f32(false, av, false, bv,
                                                (short)0, acc, false, false);
  }
  return acc;
}

// ---------------------------------------------------------------------------
// xh = x @ We^T + be        (N x 128) @ (128 x 64)
// block = 128 threads = 4 waves; wave w -> column tile [16w, 16w+16)
// ---------------------------------------------------------------------------
__global__ void __launch_bounds__(128)
xh_kernel(const float* __restrict__ x, const float* __restrict__ We,
          const float* __restrict__ be, float* __restrict__ xh, int N) {
  const int lane = threadIdx.x & 31;
  const int l15 = lane & 15, half = lane >> 4;
  const int mbase = blockIdx.x * 16;
  const int nbase = (threadIdx.x >> 5) * 16;
  const int m = min(mbase + l15, N - 1);
  v8f acc = {};
  acc = wmma_acc_bt(x, INCH, We, INCH, INCH, m, nbase + l15, half, acc);
  const float bias = be[nbase + l15];
#pragma unroll
  for (int g = 0; g < 8; ++g) {
    const int row = mbase + g + 8 * half;
    if (row < N) xh[row * HCH + nbase + l15] = acc[g] + bias;
  }
}

// ---------------------------------------------------------------------------
// xU = (1/nf) * (xh @ U^T + bU)
// ---------------------------------------------------------------------------
__global__ void __launch_bounds__(128)
xu_kernel(const float* __restrict__ xh, const float* __restrict__ U,
          const float* __restrict__ bU, const float* __restrict__ nf,
          float* __restrict__ xU, int N) {
  const int lane = threadIdx.x & 31;
  const int l15 = lane & 15, half = lane >> 4;
  const int mbase = blockIdx.x * 16;
  const int nbase = (threadIdx.x >> 5) * 16;
  const int m = min(mbase + l15, N - 1);
  v8f acc = {};
  acc = wmma_acc_bt(xh, HCH, U, HCH, HCH, m, nbase + l15, half, acc);
  const float bias = bU[nbase + l15];
#pragma unroll
  for (int g = 0; g < 8; ++g) {
    const int row = mbase + g + 8 * half;
    if (row < N) {
      const float deg = 1.0f / nf[row];   // nf in [0.1,1): no inf
      xU[row * HCH + nbase + l15] = deg * (acc[g] + bias);
    }
  }
}

// ---------------------------------------------------------------------------
// h = nf * (z @ W^T + xU)
// ---------------------------------------------------------------------------
__global__ void __launch_bounds__(128)
h_kernel(const float* __restrict__ z, const float* __restrict__ W,
         const float* __restrict__ xU, const float* __restrict__ nf,
         float* __restrict__ h, int N) {
  const int lane = threadIdx.x & 31;
  const int l15 = lane & 15, half = lane >> 4;
  const int mbase = blockIdx.x * 16;
  const int nbase = (threadIdx.x >> 5) * 16;
  const int m = min(mbase + l15, N - 1);
  v8f acc = {};
  acc = wmma_acc_bt(z, HCH, W, HCH, HCH, m, nbase + l15, half, acc);
#pragma unroll
  for (int g = 0; g < 8; ++g) {
    const int row = mbase + g + 8 * half;
    if (row < N) {
      const int idx = row * HCH + nbase + l15;
      h[idx] = nf[row] * (acc[g] + xU[idx]);
    }
  }
}

// ---------------------------------------------------------------------------
// z = (1-ALPHA)*z - ALPHA*(a @ W)          (W row-major, no transpose)
// ---------------------------------------------------------------------------
__global__ void __launch_bounds__(128)
update_kernel(const float* __restrict__ a, const float* __restrict__ W,
              float* __restrict__ z, int N) {
  const int lane = threadIdx.x & 31;
  const int l15 = lane & 15, half = lane >> 4;
  const int mbase = blockIdx.x * 16;
  const int nbase = (threadIdx.x >> 5) * 16;
  const int m = min(mbase + l15, N - 1);
  v8f acc = {};
  acc = wmma_acc_bn(a, HCH, W, HCH, HCH, m, nbase + l15, half, acc);
#pragma unroll
  for (int g = 0; g < 8; ++g) {
    const int row = mbase + g + 8 * half;
    if (row < N) {
      const int idx = row * HCH + nbase + l15;
      z[idx] = (1.0f - ALPHA) * z[idx] - ALPHA * acc[g];
    }
  }
}

// ---------------------------------------------------------------------------
// out = (nf*z + xh) @ Wlast^T + blast      (64 -> 40, masked col store)
// block = 96 threads = 3 waves -> column tiles 0,16,32
// ---------------------------------------------------------------------------
__global__ void __launch_bounds__(96)
head_kernel(const float* __restrict__ z, const float* __restrict__ xh,
            const float* __restrict__ nf, const float* __restrict__ Wl,
            const float* __restrict__ bl, float* __restrict__ out, int N) {
  const int lane = threadIdx.x & 31;
  const int l15 = lane & 15, half = lane >> 4;
  const int mbase = blockIdx.x * 16;
  const int nbase = (threadIdx.x >> 5) * 16;
  const int m = min(mbase + l15, N - 1);
  const int n = nbase + l15;
  const float nfm = nf[m];
  v8f acc = {};
  for (int k0 = 0; k0 < HCH; k0 += 4) {
    const int ka = k0 + half * 2;
    v2f av, bv;
    av.x = nfm * z[m * HCH + ka]     + xh[m * HCH + ka];
    av.y = nfm * z[m * HCH + ka + 1] + xh[m * HCH + ka + 1];
    bv.x = (n < OUTCH) ? Wl[n * HCH + ka]     : 0.0f;
    bv.y = (n < OUTCH) ? Wl[n * HCH + ka + 1] : 0.0f;
    acc = __builtin_amdgcn_wmma_f32_16x16x4_f32(false, av, false, bv,
                                                (short)0, acc, false, false);
  }
  const float bias = (n < OUTCH) ? bl[n] : 0.0f;
#pragma unroll
  for (int g = 0; g < 8; ++g) {
    const int row = mbase + g + 8 * half;
    if (row < N && n < OUTCH) out[row * OUTCH + n] = acc[g] + bias;
  }
}

// ---------------------------------------------------------------------------
// Edge phase: one wave32 per edge, 2 channels per lane.
//   m = LayerNorm(tanh(h[r]-h[c])) ; a[r] += m*nf[r] ; a[c] -= m*nf[c]
// h (12.8 MB) and a (12.8 MB) both live in the 192 MB L2: the random gathers
// and fp32 atomic scatter-adds resolve at L2 bandwidth, not HBM.
// ---------------------------------------------------------------------------
__global__ void __launch_bounds__(256)
edge_kernel(const int* __restrict__ row, const int* __restrict__ col,
            const float* __restrict__ h, const float* __restrict__ nf,
            const float* __restrict__ gamma, const float* __restrict__ beta,
            float* __restrict__ a, int E) {
  const int lane = threadIdx.x & 31;
  const int wid = (int)((blockIdx.x * blockDim.x + threadIdx.x) >> 5);
  if (wid >= E) return;
  const int r = row[wid];
  const int c = col[wid];
  const float2 hr = *(const float2*)(h + r * HCH + 2 * lane);
  const float2 hc = *(const float2*)(h + c * HCH + 2 * lane);
  float t0 = tanhf(hr.x - hc.x);
  float t1 = tanhf(hr.y - hc.y);
  // mean over 64 channels (wave32 butterfly)
  float s = t0 + t1;
#pragma unroll
  for (int off = 16; off >= 1; off >>= 1) s += __shfl_xor(s, off, 32);
  const float mu = s * (1.0f / 64.0f);
  const float d0 = t0 - mu, d1 = t1 - mu;
  float v = d0 * d0 + d1 * d1;
#pragma unroll
  for (int off = 16; off >= 1; off >>= 1) v += __shfl_xor(v, off, 32);
  const float rstd = rsqrtf(v * (1.0f / 64.0f) + LN_EPS);
  const float2 g = *(const float2*)(gamma + 2 * lane);
  const float2 b = *(const float2*)(beta + 2 * lane);
  const float m0 = g.x * d0 * rstd + b.x;
  const float m1 = g.y * d1 * rstd + b.y;
  const float nfr = nf[r], nfc = nf[c];
  float* ar = a + r * HCH + 2 * lane;
  float* ac = a + c * HCH + 2 * lane;
  unsafeAtomicAdd(ar,      m0 * nfr);   // global_atomic_add_f32, hits L2
  unsafeAtomicAdd(ar + 1,  m1 * nfr);
  unsafeAtomicAdd(ac,     -m0 * nfc);
  unsafeAtomicAdd(ac + 1, -m1 * nfc);
}

__global__ void fill0(float* __restrict__ p, int n) {
  const int i = blockIdx.x * blockDim.x + threadIdx.x;
  if (i < n) p[i] = 0.0f;
}

// ---------------------------------------------------------------------------
extern "C" void kernel_launch(void* const* d_in, const int* in_sizes, int n_in,
                              void* d_out, int out_size, void* d_ws, size_t ws_size,
                              hipStream_t stream) {
  const float* x     = (const float*)d_in[0];
  const int*   ei    = (const int*)d_in[1];
  const float* nf    = (const float*)d_in[2];
  const float* We    = (const float*)d_in[3];
  const float* be    = (const float*)d_in[4];
  const float* W     = (const float*)d_in[5];
  const float* U     = (const float*)d_in[6];
  const float* bU    = (const float*)d_in[7];
  const float* gamma = (const float*)d_in[8];
  const float* beta  = (const float*)d_in[9];
  const float* Wl    = (const float*)d_in[10];
  const float* bl    = (const float*)d_in[11];
  float* out = (float*)d_out;

  const int N = in_sizes[0] / INCH;     // 50000
  const int E = in_sizes[1] / 2;        // 1.6M
  const int* row = ei;
  const int* col = ei + E;

  float* xh = (float*)d_ws;             // N*64 each
  float* xU = xh + (size_t)N * HCH;
  float* z  = xU + (size_t)N * HCH;
  float* h  = z  + (size_t)N * HCH;
  float* a  = h  + (size_t)N * HCH;

  const int mt = (N + 15) / 16;                  // 3125 row tiles
  const int gz = (N * HCH + 255) / 256;
  const int ge = (int)(((size_t)E * 32 + 255) / 256);

  xh_kernel<<<mt, 128, 0, stream>>>(x, We, be, xh, N);
  xu_kernel<<<mt, 128, 0, stream>>>(xh, U, bU, nf, xU, N);
  fill0<<<gz, 256, 0, stream>>>(z, N * HCH);

  // 3 no-grad + 1 grad forward steps: identical math, unrolled on host
  for (int it = 0; it < 4; ++it) {
    h_kernel<<<mt, 128, 0, stream>>>(z, W, xU, nf, h, N);
    fill0<<<gz, 256, 0, stream>>>(a, N * HCH);
    edge_kernel<<<ge, 256, 0, stream>>>(row, col, h, nf, gamma, beta, a, E);
    update_kernel<<<mt, 128, 0, stream>>>(a, W, z, N);
  }

  head_kernel<<<mt, 96, 0, stream>>>(z, xh, nf, Wl, bl, out, N);
}